// GLCMExtractor_70858370449668
// MI455X (gfx1250) — compile-verified
//
#include <hip/hip_runtime.h>
#include <math.h>

// ---------------------------------------------------------------------------
// GLCM texture properties on MI455X (gfx1250)
//
// Strategy (one 1024-thread workgroup per image, 32 workgroups):
//   Phase A: zero 256KB LDS histogram (320KB/WGP on CDNA5 makes this possible)
//   Phase B: TDM (tensor_load_to_lds) DMAs 8-row / 16KB image tiles into a
//            double-buffered LDS stage; all waves bin pairs with ds_add_u32.
//            Sync via TENSORcnt + workgroup barriers.
//   Phase C: waves 0-15 compute energy = sqrt(sum C^2) as trace(C*C^T) with
//            V_WMMA_F32_16X16X4_F32 (A = B = 16-row block of the symmetrized
//            histogram); waves 16-31 accumulate the 6 scalar moments.
//   Phase D: thread 0 finalizes the 5 properties.
// ---------------------------------------------------------------------------

#define LEVELS 256
#define IMG_H 512
#define IMG_W 512
#define ROWS_PER_TILE 8
#define TILE_ELEMS (IMG_W * ROWS_PER_TILE)        // 4096 floats = 16 KB
#define NUM_TILES (IMG_H / ROWS_PER_TILE)         // 64
#define HIST_BYTES (LEVELS * LEVELS * 4)          // 262144
#define STAGE_BYTES (2 * TILE_ELEMS * 4)          // 32768
#define SMEM_BYTES (HIST_BYTES + STAGE_BYTES + 32)

typedef __attribute__((ext_vector_type(2))) float        v2f;
typedef __attribute__((ext_vector_type(8))) float        v8f;
typedef __attribute__((ext_vector_type(4))) unsigned int u32x4;
typedef __attribute__((ext_vector_type(8))) int          i32x8;
typedef __attribute__((ext_vector_type(4))) int          i32x4;

#define A_CON 0
#define A_DIS 1
#define A_HOM 2
#define A_SI  3
#define A_SII 4
#define A_SIJ 5
#define A_PP  6

// Issue a TDM load of one 512x8 fp32 tile of the image into LDS.
// D# built per CDNA5 ISA ch.8 (group0: count/lds_addr/global_addr/type=2,
// group1: data_size=4B, tensor 512x512, tile 512x8, stride0=512).
__device__ __forceinline__ void tdm_issue_tile(unsigned int lds_byte_off,
                                               const float* gptr) {
  unsigned long long ga = (unsigned long long)(uintptr_t)gptr;
  u32x4 g0 = {};
  g0.x = 1u;                                            // count = 1 valid D#
  g0.y = lds_byte_off;                                  // LDS dest (bytes)
  g0.z = (unsigned int)(ga & 0xFFFFFFFFu);              // global addr [31:0]
  g0.w = (unsigned int)((ga >> 32) & 0x01FFFFFFu)       // global addr [56:32]
       | (2u << 30);                                    // type = 2 ("image")
  i32x8 g1 = {};
  g1[0] = (int)(2u << 16);                              // data_size = 2 -> 4B
  g1[1] = (int)((IMG_W & 0xFFFFu) << 16);               // tensor_dim0 lo16
  g1[2] = (int)(((unsigned)IMG_W >> 16) |
                ((IMG_H & 0xFFFFu) << 16));             // dim0 hi16 | dim1 lo16
  g1[3] = (int)(((unsigned)IMG_H >> 16) |
                ((IMG_W & 0xFFFFu) << 16));             // dim1 hi16 | tile_dim0
  g1[4] = ROWS_PER_TILE;                                // tile_dim1=8, tile_dim2=0
  g1[5] = IMG_W;                                        // dim0_stride lo32
  g1[6] = (int)((((unsigned)(IMG_W * IMG_H)) & 0xFFFFu) << 16); // dim1_stride lo16
  g1[7] = (int)(((unsigned)(IMG_W * IMG_H)) >> 16);     // dim1_stride hi32
  i32x4 z4 = {};
  i32x8 z8 = {};
  // amdgpu-toolchain (clang-23) 6-arg form; groups 2/3 + extra group unused
  // for a 2D tensor, cpol = 0.
  __builtin_amdgcn_tensor_load_to_lds(g0, g1, z4, z4, z8, 0);
}

__device__ __forceinline__ int quant(float v) {
  int q = (int)(v * 256.0f);            // trunc-toward-zero, matches astype(i32)
  q = q < 0 ? 0 : q;
  return q > 255 ? 255 : q;
}

__global__ __launch_bounds__(1024) void glcm_props_kernel(
    const float* __restrict__ x, float* __restrict__ out) {
  extern __shared__ unsigned char smem[];
  unsigned int* hist = (unsigned int*)smem;
  float* stage = (float*)(smem + HIST_BYTES);
  float* acc   = (float*)(smem + HIST_BYTES + STAGE_BYTES);

  const int tid = threadIdx.x;
  const int img = blockIdx.x;
  const float* src = x + (size_t)img * (IMG_H * IMG_W);

  // ---- Phase A: zero histogram + accumulators -----------------------------
  {
    u32x4 zz = {};
    u32x4* h4 = (u32x4*)hist;
    for (int i = tid; i < (LEVELS * LEVELS / 4); i += 1024) h4[i] = zz;
    if (tid < 8) acc[tid] = 0.0f;
  }
  // Prefetch tile 0 via TDM (only wave 0 issues; doesn't touch hist region).
  if (tid < 32) {
    tdm_issue_tile((unsigned int)(uintptr_t)stage, src);
  }
  __syncthreads();

  // ---- Phase B: streamed histogram build ----------------------------------
  for (int t = 0; t < NUM_TILES; ++t) {
    if (tid < 32) {
      if (t + 1 < NUM_TILES) {
        unsigned int lds_off =
            (unsigned int)(uintptr_t)(stage + ((t + 1) & 1) * TILE_ELEMS);
        tdm_issue_tile(lds_off, src + (size_t)(t + 1) * TILE_ELEMS);
        __builtin_amdgcn_s_wait_tensorcnt(1);   // oldest (tile t) complete
      } else {
        __builtin_amdgcn_s_wait_tensorcnt(0);
      }
    }
    __syncthreads();                            // tile t visible to all waves

    const float* stg = stage + (t & 1) * TILE_ELEMS;
    const int r  = tid >> 7;                    // 8 rows per tile
    const int c0 = tid & 127;                   // 128 threads per row
    const float* row = stg + r * IMG_W;
    for (int c = c0; c < IMG_W - 1; c += 128) {
      int qa = quant(row[c]);
      int qb = quant(row[c + 1]);
      atomicAdd(&hist[qa * LEVELS + qb], 1u);   // ds_add_u32
    }
    __syncthreads();                            // tile consumed: buffer reusable
  }

  // ---- Phase C: reductions over the symmetrized GLCM ----------------------
  const int wave = tid >> 5;
  const int lane = tid & 31;

  if (wave >= 16) {
    // Scalar moments: C_ij = H_ij + H_ji (unnormalized P).
    const int t2 = tid - 512;
    float s_con = 0.f, s_dis = 0.f, s_hom = 0.f;
    float s_i = 0.f, s_ii = 0.f, s_ij = 0.f;
    for (int k = 0; k < 128; ++k) {
      int bin = t2 + k * 512;
      int i = bin >> 8;
      int j = bin & 255;
      float C = (float)(hist[bin] + hist[j * LEVELS + i]);
      float fi = (float)i, fj = (float)j;
      float d = fi - fj;
      float d2 = d * d;
      s_con += C * d2;
      s_dis += C * fabsf(d);
      s_hom += C / (1.0f + d2);
      s_i  += C * fi;
      s_ii += C * fi * fi;
      s_ij += C * fi * fj;
    }
    for (int off = 16; off > 0; off >>= 1) {
      s_con += __shfl_down(s_con, off, 32);
      s_dis += __shfl_down(s_dis, off, 32);
      s_hom += __shfl_down(s_hom, off, 32);
      s_i   += __shfl_down(s_i,   off, 32);
      s_ii  += __shfl_down(s_ii,  off, 32);
      s_ij  += __shfl_down(s_ij,  off, 32);
    }
    if (lane == 0) {
      atomicAdd(&acc[A_CON], s_con);
      atomicAdd(&acc[A_DIS], s_dis);
      atomicAdd(&acc[A_HOM], s_hom);
      atomicAdd(&acc[A_SI],  s_i);
      atomicAdd(&acc[A_SII], s_ii);
      atomicAdd(&acc[A_SIJ], s_ij);
    }
  } else {
    // Energy numerator: sum_ij C_ij^2 = trace(C * C^T), via WMMA Gram blocks.
    // Wave w owns rows [16w, 16w+16); A = B = that 16x256 block, K step 4.
    // A 16x4 f32 layout: lanes 0-15 -> M=lane, K={k0,k0+1};
    //                    lanes 16-31 -> M=lane-16, K={k0+2,k0+3}. B mirrors A.
    v8f dacc = {};
    const int mrow = (wave << 4) + (lane & 15);
    const int koff = (lane >> 4) << 1;
    for (int k0 = 0; k0 < LEVELS; k0 += 4) {
      int k = k0 + koff;
      v2f a;
      a.x = (float)(hist[mrow * LEVELS + k]     + hist[k * LEVELS + mrow]);
      a.y = (float)(hist[mrow * LEVELS + k + 1] + hist[(k + 1) * LEVELS + mrow]);
      dacc = __builtin_amdgcn_wmma_f32_16x16x4_f32(
          false, a, false, a, (short)0, dacc, false, false);
    }
    // Diagonal of D (16x16 f32 C/D layout): VGPR r -> (lane r: M=N=r) and
    // (lane r+24: M=N=r+8).
    float tr = 0.0f;
#pragma unroll
    for (int r2 = 0; r2 < 8; ++r2) {
      tr += __shfl(dacc[r2], r2, 32);
      tr += __shfl(dacc[r2], r2 + 24, 32);
    }
    if (lane == 0) atomicAdd(&acc[A_PP], tr);
  }
  __syncthreads();

  // ---- Phase D: finalize the 5 properties ---------------------------------
  if (tid == 0) {
    const float invS = 1.0f / (2.0f * (float)(IMG_H * (IMG_W - 1))); // 1/523264
    float con = acc[A_CON] * invS;
    float dis = acc[A_DIS] * invS;
    float hom = acc[A_HOM] * invS;
    float en  = sqrtf(acc[A_PP]) * invS;
    float mu  = acc[A_SI] * invS;                 // mu_i == mu_j (P symmetric)
    float var = acc[A_SII] * invS - mu * mu;      // var_i == var_j
    float cov = acc[A_SIJ] * invS - mu * mu;
    float corr = (var < 1e-15f) ? 1.0f : (cov / var);
    float* o = out + img * 5;
    o[0] = con; o[1] = dis; o[2] = hom; o[3] = en; o[4] = corr;
  }
}

extern "C" void kernel_launch(void* const* d_in, const int* in_sizes, int n_in,
                              void* d_out, int out_size, void* d_ws, size_t ws_size,
                              hipStream_t stream) {
  (void)in_sizes; (void)n_in; (void)d_ws; (void)ws_size; (void)out_size;
  const float* x = (const float*)d_in[0];
  float* out = (float*)d_out;
  glcm_props_kernel<<<dim3(32), dim3(1024), SMEM_BYTES, stream>>>(x, out);
}